// bb_creation_52252572123763
// MI455X (gfx1250) — compile-verified
//
#include <hip/hip_runtime.h>
#include <cstdint>
#include <cstddef>

// ---------------- problem constants (shapes are fixed by the reference) ----
#define C1     80      // classes - 1
#define NROI   4096
#define NCLS   81
#define ROW    85      // 81 class scores + 4 box coords per output row
#define RT     64      // ROIs per block tile
#define CT     8       // classes per block chunk
#define IMG_W  1216.0f
#define IMG_H  800.0f

typedef __attribute__((ext_vector_type(4))) unsigned int u32x4;
typedef __attribute__((ext_vector_type(8))) int          i32x8;
typedef __attribute__((ext_vector_type(4))) int          i32x4;
typedef __attribute__((ext_vector_type(4))) float        f32x4;

// Dynamic LDS layout (starts at LDS offset 0 -> TDM lds_addr == 0 is exact):
//   [0, RT*NCLS)                 : sc  — class-score tile (20736 B)
//   [RT*NCLS, RT*NCLS + RT*CT*4) : bxs — decoded boxes, layout [r][c][4] (8192 B)
#define SC_FLOATS  (RT * NCLS)                     // 5184
#define LDS_BYTES  ((SC_FLOATS + RT * CT * 4) * 4) // 28928

#if __has_builtin(__builtin_amdgcn_tensor_load_to_lds) && __has_builtin(__builtin_amdgcn_s_wait_tensorcnt)
#define USE_TDM 1
#else
#define USE_TDM 0
#endif

__global__ __launch_bounds__(256) void bb_decode_kernel(
    const float* __restrict__ out_regr,   // (NROI, C1*4)
    const float* __restrict__ out_class,  // (NROI, NCLS)
    const float* __restrict__ rois,       // (NROI, 4)
    float* __restrict__ out)              // (C1, NROI, ROW) flat
{
  extern __shared__ float smem[];
  float* sc  = smem;                // class scores tile
  float* bxs = smem + SC_FLOATS;    // boxes tile [r_local][c_local][4]

  const int tid = threadIdx.x;
  const int rt  = blockIdx.x & 63;  // NROI/RT == 64 tiles
  const int cc  = blockIdx.x >> 6;  // C1/CT == 10 chunks
  const int r0  = rt * RT;
  const int c0  = cc * CT;

  // ---- Phase 1: kick off TDM async copy of the 64x81 score tile into LDS --
#if USE_TDM
  if (tid < 32) {  // one wave issues the descriptor (TDM ignores EXEC)
    const uint64_t gaddr = (uint64_t)(uintptr_t)(out_class + (size_t)r0 * NCLS);
    u32x4 g0;
    g0.x = 1u;                                            // count=1 valid D#
    g0.y = 0u;                                            // lds_addr = 0 (sc)
    g0.z = (unsigned)(gaddr & 0xFFFFFFFFu);               // global_addr[31:0]
    g0.w = (unsigned)((gaddr >> 32) & 0x01FFFFFFu)        // global_addr[56:32]
         | (2u << 30);                                    // type = 2 ("image")
    i32x8 g1;
    g1[0] = 2 << 16;                 // data_size = 4B; no multicast/pad/iter
    g1[1] = NCLS << 16;              // tensor_dim0 = 81 (lo16 at bits 63:48)
    g1[2] = RT << 16;                // tensor_dim0 hi=0 | tensor_dim1 = 64
    g1[3] = NCLS << 16;              // tensor_dim1 hi=0 | tile_dim0 = 81
    g1[4] = RT;                      // tile_dim1 = 64   | tile_dim2 = 0
    g1[5] = NCLS;                    // tensor_dim0_stride = 81 (lo32)
    g1[6] = 0;                       // stride hi | tensor_dim1_stride lo (2D: unused)
    g1[7] = 0;
    const i32x4 gz4 = {0, 0, 0, 0};  // groups 2/3 unused (2D tensor)
    const i32x8 gz8 = {0, 0, 0, 0, 0, 0, 0, 0};
    __builtin_amdgcn_tensor_load_to_lds(g0, g1, gz4, gz4, gz8, 0);
  }
#else
  // Fallback: cooperative vectorized fill of the score tile.
  {
    const f32x4* src = (const f32x4*)(out_class + (size_t)r0 * NCLS);
    for (int k = tid; k < SC_FLOATS / 4; k += 256)
      ((f32x4*)sc)[k] = src[k];
  }
#endif

  // ---- Phase 2 (overlaps TDM): decode CT x RT boxes into LDS --------------
  for (int p = tid; p < CT * RT; p += 256) {
    const int r_local = p >> 3;        // 0..63
    const int c_local = p & 7;         // 0..7
    const int r = r0 + r_local;
    const int c = c0 + c_local;
    // 16B-aligned float4 load of the regression targets for (r, c)
    const f32x4 t = *(const f32x4*)(out_regr + (size_t)r * (C1 * 4) + c * 4);
    const float x = rois[r * 4 + 0];
    const float y = rois[r * 4 + 1];
    const float w = rois[r * 4 + 2];
    const float h = rois[r * 4 + 3];
    // divide by STD = {8,8,4,4} -> exact power-of-two multiplies
    const float t0 = t.x * 0.125f, t1 = t.y * 0.125f;
    const float t2 = t.z * 0.25f,  t3 = t.w * 0.25f;
    const float cx1 = t0 * w + x + 0.5f * w;
    const float cy1 = t1 * h + y + 0.5f * h;
    const float w1  = expf(t2) * w;
    const float h1  = expf(t3) * h;
    const float gx  = rintf(cx1 - 0.5f * w1);   // jnp.round == nearest-even
    const float gy  = rintf(cy1 - 0.5f * h1);
    const float gw  = rintf(w1);
    const float gh  = rintf(h1);
    f32x4 bv;
    bv.x = gx / IMG_W;
    bv.y = gy / IMG_H;
    bv.z = (gx + gw) / IMG_W;
    bv.w = (gy + gh) / IMG_H;
    *(f32x4*)(bxs + (size_t)p * 4) = bv;        // [r_local][c_local][4]
  }

  // ---- Phase 3: make the TDM tile + boxes visible to the whole block ------
#if USE_TDM
  if (tid < 32) __builtin_amdgcn_s_wait_tensorcnt(0);
#endif
  __syncthreads();

  // ---- Phase 4: stream out CT contiguous 64x85 chunks with b128 stores ----
  // Chunk base ((c*NROI + r0)*85 floats) is 16B-aligned: 64*85*4 = 21760 B.
  for (int c_local = 0; c_local < CT; ++c_local) {
    const int c = c0 + c_local;
    float* obase = out + ((size_t)c * NROI + (size_t)r0) * ROW;
    for (int k = tid; k < (RT * ROW) / 4; k += 256) {   // 1360 float4 / chunk
      const int e = k * 4;
      int r = e / ROW;            // constant-divisor -> mulhi
      int i = e - r * ROW;
      float tmp[4];
#pragma unroll
      for (int j = 0; j < 4; ++j) {
        tmp[j] = (i < NCLS) ? sc[r * NCLS + i]
                            : bxs[((r * CT) + c_local) * 4 + (i - NCLS)];
        if (++i == ROW) { i = 0; ++r; }   // increment-with-carry, no div
      }
      const f32x4 v = {tmp[0], tmp[1], tmp[2], tmp[3]};
      *(f32x4*)(obase + e) = v;
    }
  }
}

extern "C" void kernel_launch(void* const* d_in, const int* in_sizes, int n_in,
                              void* d_out, int out_size, void* d_ws, size_t ws_size,
                              hipStream_t stream) {
  (void)in_sizes; (void)n_in; (void)d_ws; (void)ws_size; (void)out_size;
  const float* out_regr  = (const float*)d_in[0];  // (1, 4096, 320) f32
  const float* out_class = (const float*)d_in[1];  // (1, 4096, 81)  f32
  const float* rois      = (const float*)d_in[2];  // (1, 4096, 4)   f32
  // d_in[3] ("b") only contributes its shape: H=800, W=1216 (compile-time).
  float* out = (float*)d_out;                      // (80, 1, 4096, 85) f32

  const int grid = (NROI / RT) * (C1 / CT);        // 64 * 10 = 640 blocks
  bb_decode_kernel<<<grid, 256, LDS_BYTES, stream>>>(out_regr, out_class, rois, out);
}